// GumbelSinkhorn_87565793230958
// MI455X (gfx1250) — compile-verified
//
#include <hip/hip_runtime.h>
#include <hip/hip_bf16.h>
#include <cstdint>

#ifndef __has_builtin
#define __has_builtin(x) 0
#endif

#define GS_NEG     (-1.0e30f)
#define GS_AT      256          // agents (rows) = tasks (cols) = 256
#define GS_STRIDE  258          // LDS row stride in floats: bank-skew (258%64==2) + 8B align
#define GS_ITER    5
#define GS_LDS_BYTES (GS_AT * GS_STRIDE * 4)

typedef __attribute__((address_space(1))) int GInt;   // global (prints as __device__ int*)
typedef __attribute__((address_space(3))) int LInt;   // LDS    (prints as __shared__ int*)

__global__ void __launch_bounds__(256, 1)
gumbel_sinkhorn_kernel(const float* __restrict__ logits,
                       const int* __restrict__ free_agents_num,
                       const int* __restrict__ tasks_num,
                       float* __restrict__ out)
{
    extern __shared__ float tile[];             // [256][258] floats, 258-KB tile
    const int b = blockIdx.x;
    const int t = threadIdx.x;                  // 0..255
    const int A = free_agents_num[b];
    const int T = tasks_num[b];
    const size_t base = (size_t)b * (GS_AT * GS_AT);
    const float* src = logits + base + t;

    // ---- stage the whole batch tile into LDS (CDNA5 async global->LDS DMA) ----
#if __has_builtin(__builtin_amdgcn_global_load_async_to_lds_b32)
    #pragma unroll 4
    for (int k = 0; k < GS_AT; ++k) {
        __builtin_amdgcn_global_load_async_to_lds_b32(
            (GInt*)(src + k * GS_AT),
            (LInt*)(tile + k * GS_STRIDE + t),
            0, 0);
    }
  #if __has_builtin(__builtin_amdgcn_s_wait_asynccnt)
    __builtin_amdgcn_s_wait_asynccnt(0);
  #else
    asm volatile("s_wait_asynccnt 0" ::: "memory");
  #endif
#else
    #pragma unroll 4
    for (int k = 0; k < GS_AT; ++k)
        tile[k * GS_STRIDE + t] = src[k * GS_AT];
#endif
    __syncthreads();

    // ---- apply validity mask: invalid entries -> -1e30 (stays exact through fp32 subtracts)
    #pragma unroll 4
    for (int k = 0; k < GS_AT; ++k)
        if (k >= A || t >= T)
            tile[k * GS_STRIDE + t] = GS_NEG;
    __syncthreads();

    for (int it = 0; it < GS_ITER; ++it) {
        // ---------- row normalize (logsumexp over tasks axis); thread t owns row t ----------
        {
            const int r = t;
            if (r < A && T > 0) {               // skip fully-masked rows (reference re-masks them)
                float2* row = (float2*)(tile + r * GS_STRIDE);  // 8B-aligned: r*1032
                float m0 = GS_NEG, m1 = GS_NEG;
                #pragma unroll 8
                for (int j = 0; j < GS_AT / 2; ++j) {
                    float2 v = row[j];
                    m0 = fmaxf(m0, v.x); m1 = fmaxf(m1, v.y);
                }
                const float m = fmaxf(m0, m1);
                float s0 = 0.f, s1 = 0.f;
                #pragma unroll 8
                for (int j = 0; j < GS_AT / 2; ++j) {
                    float2 v = row[j];
                    s0 += __expf(v.x - m); s1 += __expf(v.y - m);  // masked cols underflow to 0
                }
                const float lse = m + __logf(s0 + s1);
                #pragma unroll 8
                for (int j = 0; j < GS_AT / 2; ++j) {
                    float2 v = row[j]; v.x -= lse; v.y -= lse; row[j] = v;
                }
            }
        }
        __syncthreads();

        // ---------- col normalize (logsumexp over agents axis); thread t owns column t ----------
        const int c = t;
        const bool colActive = (c < T) && (A > 0);
        if (it < GS_ITER - 1) {
            if (colActive) {
                float m = GS_NEG;
                #pragma unroll 8
                for (int r = 0; r < GS_AT; ++r)
                    m = fmaxf(m, tile[r * GS_STRIDE + c]);
                float s = 0.f;
                #pragma unroll 8
                for (int r = 0; r < GS_AT; ++r)
                    s += __expf(tile[r * GS_STRIDE + c] - m);
                const float lse = m + __logf(s);
                #pragma unroll 8
                for (int r = 0; r < GS_AT; ++r)
                    tile[r * GS_STRIDE + c] -= lse;
            }
            __syncthreads();
        } else {
            // ---- last col pass fused with output: exp(x - lse + 1e-6), coalesced store ----
            float lse = 0.f;
            if (colActive) {
                float m = GS_NEG;
                #pragma unroll 8
                for (int r = 0; r < GS_AT; ++r)
                    m = fmaxf(m, tile[r * GS_STRIDE + c]);
                float s = 0.f;
                #pragma unroll 8
                for (int r = 0; r < GS_AT; ++r)
                    s += __expf(tile[r * GS_STRIDE + c] - m);
                lse = m + __logf(s);
            }
            float* dst = out + base + c;
            #pragma unroll 4
            for (int r = 0; r < GS_AT; ++r) {
                float v = 0.f;
                if (colActive && r < A)
                    v = __expf(tile[r * GS_STRIDE + c] - lse + 1e-6f);
                dst[r * GS_AT] = v;             // (b, r, c): contiguous across threads
            }
        }
    }
}

extern "C" void kernel_launch(void* const* d_in, const int* in_sizes, int n_in,
                              void* d_out, int out_size, void* d_ws, size_t ws_size,
                              hipStream_t stream) {
    (void)n_in; (void)out_size; (void)d_ws; (void)ws_size;
    const float* logits = (const float*)d_in[0];
    const int*   freeA  = (const int*)d_in[1];
    const int*   tasks  = (const int*)d_in[2];
    float*       out    = (float*)d_out;
    const int B = in_sizes[1];                  // 512 batches

    (void)hipFuncSetAttribute(reinterpret_cast<const void*>(gumbel_sinkhorn_kernel),
                              hipFuncAttributeMaxDynamicSharedMemorySize, GS_LDS_BYTES);
    gumbel_sinkhorn_kernel<<<B, 256, GS_LDS_BYTES, stream>>>(logits, freeA, tasks, out);
}